// ConstructionEmbedding_25099788878675
// MI455X (gfx1250) — compile-verified
//
#include <hip/hip_runtime.h>

typedef __attribute__((ext_vector_type(2))) float v2f;
typedef __attribute__((ext_vector_type(8))) float v8f;

namespace {
constexpr int NB   = 256;   // batch
constexpr int NN   = 5000;  // nodes per batch
constexpr int NK   = 100;   // candidates
constexpr int DIM  = 128;   // embedding dim
constexpr int ROWS = NK + 2;
}

// Kernel 1: gathered coord embedding (Linear 2->128) + candidate compaction scatter.
// One block per batch, one thread per embedding channel.
__global__ void __launch_bounds__(DIM)
coord_embed_gather(const float* __restrict__ nodes,
                   const int*  __restrict__ first_idx,
                   const int*  __restrict__ last_idx,
                   const int*  __restrict__ cand,
                   const float* __restrict__ cW,
                   const float* __restrict__ cb,
                   float* __restrict__ X1,
                   float* __restrict__ X2,
                   float* __restrict__ out)
{
  const int b = blockIdx.x;
  const int d = threadIdx.x;            // 0..127 channel

  __shared__ int s_idx[NK];
  __shared__ int s_pos[NK];
  __shared__ int s_count;

  if (d < NK) s_idx[d] = cand[b * NK + d];
  __syncthreads();
  if (d < NK) {
    int p = 0;
    for (int j = 0; j < d; ++j) p += (s_idx[j] != -1);
    s_pos[d] = p;                        // exclusive prefix over valid mask
  }
  if (d == 0) {
    int c = 0;
    for (int j = 0; j < NK; ++j) c += (s_idx[j] != -1);
    s_count = c;
  }
  __syncthreads();

  const float w0   = cW[d];
  const float w1   = cW[DIM + d];
  const float bias = cb[d];
  const float* nb  = nodes + (size_t)b * NN * 2;
  float* ob        = out + (size_t)b * ROWS * DIM;

  {
    const int fi = first_idx[b];
    X1[b * DIM + d] = fmaf(nb[fi * 2 + 0], w0, fmaf(nb[fi * 2 + 1], w1, bias));
    const int li = last_idx[b];
    X2[b * DIM + d] = fmaf(nb[li * 2 + 0], w0, fmaf(nb[li * 2 + 1], w1, bias));
  }

  // compacted candidate embeddings (rows 2..), zeros for dropped tail slots
  for (int j = 0; j < NK; ++j) {
    const int ci = s_idx[j];
    if (ci != -1) {
      const float e = fmaf(nb[ci * 2 + 0], w0, fmaf(nb[ci * 2 + 1], w1, bias));
      ob[(2 + s_pos[j]) * DIM + d] = e;
    }
  }
  for (int s = s_count; s < NK; ++s) ob[(2 + s) * DIM + d] = 0.0f;
}

// Kernel 2: fp32 WMMA heads: out[:,0,:] = X1 @ W1 + b1 ; out[:,1,:] = X2 @ W2 + b2
// grid = 16 M-tiles; block = 256 threads = 8 wave32; wave w owns N-tile w.
__global__ void __launch_bounds__(256)
wmma_heads(const float* __restrict__ X1, const float* __restrict__ X2,
           const float* __restrict__ W1, const float* __restrict__ b1,
           const float* __restrict__ W2, const float* __restrict__ b2,
           float* __restrict__ out)
{
  const int lane  = threadIdx.x & 31;
  const int wave  = threadIdx.x >> 5;   // 0..7  -> N tile
  const int mtile = blockIdx.x;         // 0..15 -> 16 batches
  const int r  = lane & 15;             // A: M row; B/D: N column within tile
  const int hi = lane >> 4;             // K sub-block select (A/B), +8 M rows (D)
  const int n  = wave * 16 + r;         // global output column

  const float* Xg[2] = { X1, X2 };
  const float* Wg[2] = { W1, W2 };
  const float* Bg[2] = { b1, b2 };

  #pragma unroll
  for (int g = 0; g < 2; ++g) {
    const float* X = Xg[g];
    const float* W = Wg[g];
    v8f c = {};
    #pragma unroll
    for (int kk = 0; kk < DIM / 4; ++kk) {
      const int k = kk * 4;
      // A 16x4 f32: lane holds X[row, k+2*hi .. k+2*hi+1] (8B-aligned pair)
      const v2f a = *(const v2f*)(X + (mtile * 16 + r) * DIM + k + 2 * hi);
      // B 4x16 f32: lane holds W[k+2*hi, n], W[k+2*hi+1, n]
      v2f bb;
      bb.x = W[(k + 2 * hi + 0) * DIM + n];
      bb.y = W[(k + 2 * hi + 1) * DIM + n];
      c = __builtin_amdgcn_wmma_f32_16x16x4_f32(
              /*neg_a=*/false, a, /*neg_b=*/false, bb,
              /*c_mod=*/(short)0, c, /*reuse_a=*/false, /*reuse_b=*/false);
    }
    const float bias = Bg[g][n];
    #pragma unroll
    for (int i = 0; i < 8; ++i) {
      const int batch = mtile * 16 + i + 8 * hi;   // D: VGPR i -> M = i + 8*hi
      out[(size_t)batch * ROWS * DIM + g * DIM + n] = c[i] + bias;
    }
  }
}

extern "C" void kernel_launch(void* const* d_in, const int* in_sizes, int n_in,
                              void* d_out, int out_size, void* d_ws, size_t ws_size,
                              hipStream_t stream)
{
  const float* nodes     = (const float*)d_in[0];
  const int*   first_idx = (const int*)d_in[1];
  const int*   last_idx  = (const int*)d_in[2];
  const int*   cand      = (const int*)d_in[3];
  const float* cW        = (const float*)d_in[4];
  const float* cb        = (const float*)d_in[5];
  const float* W1        = (const float*)d_in[6];
  const float* b1        = (const float*)d_in[7];
  const float* W2        = (const float*)d_in[8];
  const float* b2        = (const float*)d_in[9];
  float* out = (float*)d_out;

  float* X1 = (float*)d_ws;             // [256,128] staged first-node embeddings
  float* X2 = X1 + NB * DIM;            // [256,128] staged last-node embeddings

  coord_embed_gather<<<NB, DIM, 0, stream>>>(nodes, first_idx, last_idx, cand,
                                             cW, cb, X1, X2, out);
  wmma_heads<<<NB / 16, 256, 0, stream>>>(X1, X2, W1, b1, W2, b2, out);
}